// Probe_59734405153028
// MI455X (gfx1250) — compile-verified
//
#include <hip/hip_runtime.h>

// ---------------------------------------------------------------------------
// Biquad filter bank (64 ch, fwd+bwd) via block state-space decomposition.
//   y[t] = b0*x[t] - a1*y[t-1] - a2*y[t-2]
// Per 16-sample block:  y = T_c @ (b0*x_blk)  +  alpha*y(-1) + beta*y(-2)
// The 16x16 intra-block convolution runs on 4 chained v_wmma_f32_16x16x4_f32
// per 256-sample tile; cross-block state is a 2-state f64 scan (512-way
// parallel). Phase-3 is branch-free around the WMMAs; all hot-path global
// traffic is 128-bit vectorized (fast path for full chunks, scalar guarded
// path only for the single partial tile-group per sequence).
// ---------------------------------------------------------------------------

#define T_LEN     120000
#define NCH       64
#define NSEQ      512        // 4 batch * 2 dir * 64 ch
#define NBLK      7500       // T_LEN / 16
#define NBLKP     7552       // padded per-seq stride (= 472 tiles)
#define NTILES    469        // ceil(NBLK / 16)
#define TG_TILES  8          // tiles per workgroup (8 waves)
#define NTG       59         // ceil(NTILES / 8)
#define CHUNK     (TG_TILES * 256)   // 2048 samples per workgroup

typedef __attribute__((ext_vector_type(2))) float v2f;
typedef __attribute__((ext_vector_type(4))) float v4f;
typedef __attribute__((ext_vector_type(8))) float v8f;

// -------------------- phase 0: per-channel tables (f64) ---------------------
__global__ void bq_setup(const double* __restrict__ aC, const double* __restrict__ bC,
                         float* __restrict__ bhp32, float* __restrict__ al32,
                         float* __restrict__ be32, double* __restrict__ mD) {
    int c = threadIdx.x;
    if (c >= NCH) return;
    double a0 = aC[c * 3 + 0];
    double a1 = aC[c * 3 + 1] / a0;
    double a2 = aC[c * 3 + 2] / a0;
    double b0 = bC[c * 3 + 0] / a0;

    double h1 = 0.0, h2 = 0.0;          // h[k-1], h[k-2]
    double al1 = 1.0, al2 = 0.0;        // alpha[-1]=1, alpha[-2]=0
    double be1 = 0.0, be2 = 1.0;        // beta[-1]=0,  beta[-2]=1
    double a14 = 0.0, a15 = 0.0, b14 = 0.0, b15 = 0.0;
    for (int k = 0; k < 16; ++k)        // zero-pad low half of Toeplitz table
        bhp32[c * 32 + k] = 0.0f;
    for (int k = 0; k < 16; ++k) {
        double imp = (k == 0) ? 1.0 : 0.0;
        double h  = imp - a1 * h1 - a2 * h2;
        double al = -a1 * al1 - a2 * al2;
        double be = -a1 * be1 - a2 * be2;
        bhp32[c * 32 + 16 + k] = (float)(b0 * h);
        al32[c * 16 + k] = (float)al;
        be32[c * 16 + k] = (float)be;
        if (k == 14) { a14 = al; b14 = be; }
        if (k == 15) { a15 = al; b15 = be; }
        h2 = h1; h1 = h; al2 = al1; al1 = al; be2 = be1; be1 = be;
    }
    mD[c * 4 + 0] = a15;   // y1' = p15 + a15*y1 + b15*y2
    mD[c * 4 + 1] = b15;
    mD[c * 4 + 2] = a14;   // y2' = p14 + a14*y1 + b14*y2
    mD[c * 4 + 3] = b14;
}

// ------ phase 1: per-block particular tail (p[14], p[15]), fully parallel ---
__global__ void bq_partsol(const float* __restrict__ x, const double* __restrict__ aC,
                           const double* __restrict__ bC,
                           float* __restrict__ P15, float* __restrict__ P14) {
    long idx = (long)blockIdx.x * 256 + threadIdx.x;
    if (idx >= (long)NSEQ * NBLK) return;
    int seq = (int)(idx / NBLK);
    int blk = (int)(idx % NBLK);
    int c = seq & 63, d = (seq >> 6) & 1, b = seq >> 7;

    double a0 = aC[c * 3 + 0];
    double a1 = aC[c * 3 + 1] / a0;
    double a2 = aC[c * 3 + 2] / a0;
    double b0 = bC[c * 3 + 0] / a0;
    const float* xb = x + (size_t)b * T_LEN;

    double p1 = 0.0, p2 = 0.0;
    int t0 = blk * 16;
#pragma unroll
    for (int i = 0; i < 16; ++i) {
        int t = t0 + i;
        int g = d ? (T_LEN - 1 - t) : t;
        double p = b0 * (double)xb[g] - a1 * p1 - a2 * p2;
        p2 = p1; p1 = p;
    }
    P15[idx] = (float)p1;   // p[15]
    P14[idx] = (float)p2;   // p[14]
}

// -------- phase 2: f64 boundary-state scan, one thread per sequence ---------
__global__ void bq_scan(const double* __restrict__ mD,
                        const float* __restrict__ P15, const float* __restrict__ P14,
                        float* __restrict__ Y1f, float* __restrict__ Y2f) {
    int seq = blockIdx.x * blockDim.x + threadIdx.x;
    if (seq >= NSEQ) return;
    int c = seq & 63;
    double m11 = mD[c * 4 + 0], m12 = mD[c * 4 + 1];
    double m21 = mD[c * 4 + 2], m22 = mD[c * 4 + 3];
    size_t pbase = (size_t)seq * NBLK;
    size_t ybase = (size_t)seq * NBLKP;
    double y1 = 0.0, y2 = 0.0;
    for (int blk = 0; blk < NBLK; ++blk) {
        Y1f[ybase + blk] = (float)y1;       // state at block start
        Y2f[ybase + blk] = (float)y2;
        double n1 = (double)P15[pbase + blk] + m11 * y1 + m12 * y2;
        double n2 = (double)P14[pbase + blk] + m21 * y1 + m22 * y2;
        y1 = n1; y2 = n2;
    }
}

__device__ __forceinline__ v4f bq_rev4(v4f v) {
    v4f r; r[0] = v[3]; r[1] = v[2]; r[2] = v[1]; r[3] = v[0]; return r;
}

// -------- phase 3: WMMA tile matmul + state correction + vector I/O ---------
__global__ void bq_wmma(const float* __restrict__ x,
                        const float* __restrict__ bhp32,
                        const float* __restrict__ al32, const float* __restrict__ be32,
                        const float* __restrict__ Y1f, const float* __restrict__ Y2f,
                        float* __restrict__ out) {
    __shared__ float xs[CHUNK];              // staged input chunk (8 KB)
    __shared__ float ys[CHUNK];              // output transpose buffer (8 KB)
    __shared__ float tabA[32];               // zero-padded Toeplitz column
    __shared__ float tabAl[16];              // alpha homogeneous solution
    __shared__ float tabBe[16];              // beta homogeneous solution

    int seq = blockIdx.x / NTG;
    int tg  = blockIdx.x % NTG;
    int c = seq & 63, d = (seq >> 6) & 1, b = seq >> 7;
    int tid = threadIdx.x;
    const float* xb = x + (size_t)b * T_LEN;

    // Stage per-channel tables into LDS.
    if (tid < 32)       tabA[tid]       = bhp32[c * 32 + tid];
    else if (tid < 48)  tabAl[tid - 32] = al32[c * 16 + tid - 32];
    else if (tid < 64)  tabBe[tid - 48] = be32[c * 16 + tid - 48];

    int  tbase = tg * CHUNK;
    bool full  = (tbase + CHUNK <= T_LEN);   // block-uniform

    // ---- Stage the 2048-sample chunk into LDS (flip applied here) ----------
    if (full) {
        v4f* dst = (v4f*)xs;
        if (d == 0) {
            const v4f* src = (const v4f*)(xb + tbase);
            dst[tid * 2 + 0] = src[tid * 2 + 0];
            dst[tid * 2 + 1] = src[tid * 2 + 1];
        } else {
            // idx = tid*8+k  <->  g = (T-1-tbase-tid*8) - k : reversed groups
            const v4f* src = (const v4f*)(xb + (T_LEN - tbase - tid * 8 - 8));
            v4f a = src[0];                  // g ascending: idx k = 7..4
            v4f bv = src[1];                 // idx k = 3..0
            dst[tid * 2 + 0] = bq_rev4(bv);
            dst[tid * 2 + 1] = bq_rev4(a);
        }
    } else {
        // Partial (last) chunk: clamped unconditional load + select.
#pragma unroll
        for (int k = 0; k < 8; ++k) {
            int idx = k * 256 + tid;
            int t = tbase + idx;
            int tc = (t < T_LEN) ? t : (T_LEN - 1);
            int g  = d ? (T_LEN - 1 - tc) : tc;
            float v = xb[g];
            xs[idx] = (t < T_LEN) ? v : 0.0f;
        }
    }
    __syncthreads();

    int w = tid >> 5, lane = tid & 31;
    int half = lane >> 4, lrow = lane & 15;  // A: row M; B/D: column N

    // D(16x16) = T_c(16x16) @ X_tile(16x16), K=16 as 4 chunks of 4.
    v8f acc = {};
#pragma unroll
    for (int kk = 0; kk < 4; ++kk) {
        int col0 = kk * 4 + half * 2;
        v2f a = { tabA[16 + lrow - col0], tabA[15 + lrow - col0] };
        v2f bb = *(const v2f*)&xs[w * 256 + lrow * 16 + col0];
        acc = __builtin_amdgcn_wmma_f32_16x16x4_f32(
            false, a, false, bb, (short)0, acc, false, false);
    }

    // State correction; Y arrays padded to NBLKP so reads are unguarded.
    int n   = lrow;
    int blk = (tg * TG_TILES + w) * 16 + n;
    float y1 = Y1f[(size_t)seq * NBLKP + blk];
    float y2 = Y2f[(size_t)seq * NBLKP + blk];
#pragma unroll
    for (int j = 0; j < 8; ++j) {
        int i = j + half * 8;                       // sample within block
        ys[w * 256 + n * 16 + i] = acc[j] + tabAl[i] * y1 + tabBe[i] * y2;
    }
    __syncthreads();

    // ---- Store the chunk (reversed for bwd), 128-bit fast path -------------
    size_t outBase = (size_t)(b * 128 + d * 64 + c) * T_LEN;
    if (full) {
        const v4f* src = (const v4f*)ys;
        if (d == 0) {
            v4f* dst = (v4f*)(out + outBase + tbase);
            dst[tid * 2 + 0] = src[tid * 2 + 0];
            dst[tid * 2 + 1] = src[tid * 2 + 1];
        } else {
            v4f* dst = (v4f*)(out + outBase + (T_LEN - tbase - tid * 8 - 8));
            v4f lo = src[tid * 2 + 0];       // ys idx k = 0..3
            v4f hi = src[tid * 2 + 1];       // ys idx k = 4..7
            dst[0] = bq_rev4(hi);            // out j=0..3 <- idx k=7..4
            dst[1] = bq_rev4(lo);            // out j=4..7 <- idx k=3..0
        }
    } else {
#pragma unroll
        for (int k = 0; k < 8; ++k) {
            int idx = k * 256 + tid;
            int t = tbase + idx;
            if (t < T_LEN) {
                int tglob = d ? (T_LEN - 1 - t) : t;
                out[outBase + tglob] = ys[idx];
            }
        }
    }
}

// ---------------------------------------------------------------------------
extern "C" void kernel_launch(void* const* d_in, const int* in_sizes, int n_in,
                              void* d_out, int out_size, void* d_ws, size_t ws_size,
                              hipStream_t stream) {
    const float*  x  = (const float*)d_in[0];   // [4,1,120000] f32
    const double* aC = (const double*)d_in[1];  // [64,3] f64
    const double* bC = (const double*)d_in[2];  // [64,3] f64
    float* out = (float*)d_out;                 // [4,128,120000] f32

    // Workspace layout (256-aligned; total ~61.7 MB)
    char* ws = (char*)d_ws;
    double* mD    = (double*)(ws + 0);                    //  2 KB
    float*  bhp32 = (float*)(ws + 2048);                  //  8 KB (padded Toeplitz)
    float*  al32  = (float*)(ws + 10240);                 //  4 KB
    float*  be32  = (float*)(ws + 14336);                 //  4 KB
    size_t  pbytes = (size_t)NSEQ * NBLK  * sizeof(float); // 15,360,000 B
    size_t  ybytes = (size_t)NSEQ * NBLKP * sizeof(float); // 15,466,496 B
    float*  P15 = (float*)(ws + 18432);
    float*  P14 = (float*)(ws + 18432 + pbytes);
    float*  Y1f = (float*)(ws + 18432 + 2 * pbytes);
    float*  Y2f = (float*)(ws + 18432 + 2 * pbytes + ybytes);
    (void)in_sizes; (void)n_in; (void)out_size; (void)ws_size;

    bq_setup<<<1, 64, 0, stream>>>(aC, bC, bhp32, al32, be32, mD);

    long nwork = (long)NSEQ * NBLK;                       // 3,840,000
    bq_partsol<<<(int)(nwork / 256), 256, 0, stream>>>(x, aC, bC, P15, P14);

    bq_scan<<<16, 32, 0, stream>>>(mD, P15, P14, Y1f, Y2f);

    bq_wmma<<<NSEQ * NTG, 256, 0, stream>>>(x, bhp32, al32, be32, Y1f, Y2f, out);
}